// BSplineBasis_35828617183709
// MI455X (gfx1250) — compile-verified
//
#include <hip/hip_runtime.h>

// B-spline basis (Cox-de Boor), degree 3, 64 control points.
// Output [B, 64] f32 -> 256 MB stream; memory-bound on stores.
// Strategy: banded 4-nonzero recurrence per point, wave-private LDS tile,
// async LDS->global DMA stores (CDNA5 GLOBAL_STORE_ASYNC_FROM_LDS_B128).

#define DEGP 3
#define NKNOTS 68          // m + p + 1 = 64 + 3 + 1
#define MCOLS 64           // output columns
#define EPSV 1e-8f

#define WAVES_PER_BLOCK 4
#define BLOCK_THREADS (WAVES_PER_BLOCK * 32)
#define ROWF 68            // padded row stride (floats): 68 dwords -> bank stride 4, only 2-phase b128 access
#define KN_OFF 0           // 68 floats
#define INV1_OFF 72        // 3*66 floats (k=1..3)
#define INV2_OFF 272       // 3*66 floats
#define TILE_OFF 480       // 16B aligned (480*4 = 1920)
#define SMEM_FLOATS (TILE_OFF + WAVES_PER_BLOCK * 32 * ROWF)  // 9184 floats = 36736 B

#if defined(__has_builtin)
#  if __has_builtin(__builtin_amdgcn_global_store_async_from_lds_b128)
#    define HAVE_ASYNC_STORE 1
#  endif
#endif
#ifndef HAVE_ASYNC_STORE
#  define HAVE_ASYNC_STORE 0
#endif

typedef int v4i __attribute__((ext_vector_type(4)));
typedef __attribute__((address_space(1))) v4i v4i_g;   // global (AS1)
typedef __attribute__((address_space(3))) v4i v4i_l;   // LDS (AS3)

__global__ __launch_bounds__(BLOCK_THREADS) void
bspline_basis_kernel(const float* __restrict__ t01,
                     const float* __restrict__ knots,
                     float* __restrict__ out, int B)
{
    __shared__ float smem[SMEM_FLOATS];
    const int tid = threadIdx.x;

    // ---- per-block precompute (reads global knots; no LDS dependency) ----
    if (tid < NKNOTS) smem[KN_OFF + tid] = knots[tid];
    for (int idx = tid; idx < 3 * 66; idx += BLOCK_THREADS) {
        const int k = idx / 66 + 1;           // 1..3
        const int c = idx - (k - 1) * 66;     // 0..65
        float i1 = 0.0f, i2 = 0.0f;
        if (c < 67 - k) {                     // nk = 67 - k valid columns
            const float d1 = knots[c + k] - knots[c];
            const float d2 = knots[k + 1 + c] - knots[1 + c];
            if (d1 > EPSV) i1 = 1.0f / d1;    // reference: where(d>EPS, ./d, 0)
            if (d2 > EPSV) i2 = 1.0f / d2;
        }
        smem[INV1_OFF + idx] = i1;
        smem[INV2_OFF + idx] = i2;
    }
    __syncthreads();

    const int w    = tid >> 5;
    const int lane = tid & 31;
    const long long waveBase = (long long)blockIdx.x * BLOCK_THREADS + (long long)w * 32;
    if (waveBase >= (long long)B) return;
    const int rem = (int)((long long)B - waveBase);
    const int n   = rem < 32 ? rem : 32;      // active points in this wave (32 for B%32==0)

    const float* kn   = smem + KN_OFF;
    const float* inv1 = smem + INV1_OFF;
    const float* inv2 = smem + INV2_OFF;
    float* rowBase = smem + TILE_OFF + (w * 32 + lane) * ROWF;

    if (lane < n) {
        const float x = t01[(size_t)(waveBase + lane)];

        // Span s = largest j in [3,63] with kn[j] <= x (monotone knots; exact
        // comparison semantics vs reference for x in [0,1)).
        int lo = 3, hi = 63;
        #pragma unroll
        for (int it = 0; it < 6; ++it) {
            const int mid = (lo + hi + 1) >> 1;
            if (kn[mid] <= x) lo = mid; else hi = mid - 1;
        }
        const int s = lo;

        // Banded Cox-de Boor: nonzeros at columns s-k .. s after level k.
        // w1(k,c) = (x - kn[c]) * inv1[k][c];  w2(k,c) = (kn[k+1+c] - x) * inv2[k][c]
        #define W1(k, c) ((x - kn[(c)]) * inv1[((k) - 1) * 66 + (c)])
        #define W2(k, c) ((kn[(k) + 1 + (c)] - x) * inv2[((k) - 1) * 66 + (c)])
        // k = 1 (N0[s] = 1)
        const float a0 = W2(1, s - 1);
        const float a1 = W1(1, s);
        // k = 2
        const float b0 = W2(2, s - 2) * a0;
        const float b1 = W1(2, s - 1) * a0 + W2(2, s - 1) * a1;
        const float b2 = W1(2, s) * a1;
        // k = 3
        const float r0 = W2(3, s - 3) * b0;
        const float r1 = W1(3, s - 2) * b0 + W2(3, s - 2) * b1;
        const float r2 = W1(3, s - 1) * b1 + W2(3, s - 1) * b2;
        const float r3 = W1(3, s) * b2;
        #undef W1
        #undef W2

        // Zero the 64-float row with b128 stores, then scatter the 4 nonzeros.
        const float4 z = make_float4(0.0f, 0.0f, 0.0f, 0.0f);
        float4* row4 = (float4*)rowBase;
        #pragma unroll
        for (int i = 0; i < 16; ++i) row4[i] = z;
        asm volatile("" ::: "memory");  // keep scatter after zeros (DS is in-order per wave)
        const int c0 = s - 3;           // in [0, 60]
        rowBase[c0 + 0] = r0;
        rowBase[c0 + 1] = r1;
        rowBase[c0 + 2] = r2;
        rowBase[c0 + 3] = r3;
    }

    // Ensure the wave's LDS tile is complete before the DMA engine reads it.
    asm volatile("s_wait_dscnt 0" ::: "memory");

    float*       gwave   = out + (size_t)waveBase * MCOLS;       // 8 KB contiguous slab
    const float* ldsWave = smem + TILE_OFF + w * 32 * ROWF;

#if HAVE_ASYNC_STORE
    if (n == 32) {
        // 16 async b128 stores per wave: each moves 32 lanes x 16 B = 512 B
        // contiguous global, remapped from the padded LDS rows. ASYNCcnt is
        // drained by S_ENDPGM's implicit wait-idle.
        #pragma unroll
        for (int t = 0; t < 16; ++t) {
            const int linear = t * 32 + lane;            // 0..511 float4 chunks
            const int pt = linear >> 4;                  // point row in tile
            const int cq = linear & 15;                  // float4 chunk in row
            const float* lsrc = ldsWave + pt * ROWF + cq * 4;
            float*       gdst = gwave + linear * 4;
            __builtin_amdgcn_global_store_async_from_lds_b128(
                (v4i_g*)gdst, (v4i_l*)lsrc, 0, 0);
        }
    } else
#endif
    {
        // Fallback / partial-wave path: coalesced b128 copies through VGPRs.
        for (int linear = lane; linear < n * 16; linear += 32) {
            const int pt = linear >> 4;
            const int cq = linear & 15;
            const float4 val = *(const float4*)(ldsWave + pt * ROWF + cq * 4);
            *(float4*)(gwave + linear * 4) = val;
        }
    }
}

extern "C" void kernel_launch(void* const* d_in, const int* in_sizes, int n_in,
                              void* d_out, int out_size, void* d_ws, size_t ws_size,
                              hipStream_t stream) {
    const float* t01   = (const float*)d_in[0];
    const float* knots = (const float*)d_in[1];
    float*       out   = (float*)d_out;
    const int B = in_sizes[0];   // t01 is [B,1]
    const int grid = (B + BLOCK_THREADS - 1) / BLOCK_THREADS;
    bspline_basis_kernel<<<grid, BLOCK_THREADS, 0, stream>>>(t01, knots, out, B);
}